// HMSANetwork_80427557585567
// MI455X (gfx1250) — compile-verified
//
#include <hip/hip_runtime.h>
#include <math.h>

// ---------------------------------------------------------------------------
// MI455X (gfx1250) fused HMSA network.
//  - f32 WMMA (V_WMMA_F32_16X16X4_F32) for all GEMMs (inputs are f32; top-k
//    threshold compare is precision-sensitive, so no downcast).
//  - scores kept in LDS (16 rows x 4113 f32 = 263KB of the 320KB WGP LDS),
//    never spilled to HBM; K/V (2.1MB each) are L2-resident.
//  - exact 819-th-largest selection via 4x8-bit radix select on monotonic
//    float keys -> identical ">= threshold" semantics to the reference.
// ---------------------------------------------------------------------------

typedef __attribute__((ext_vector_type(2))) float v2f;
typedef __attribute__((ext_vector_type(8))) float v8f;

#define NN     4096
#define DD     128
#define HH     4
#define DHD    32
#define MT     4098          // N + 2 anchors
#define MP     4112          // padded to 257*16
#define SCS    4113          // LDS row stride (odd vs 64 banks)
#define KKEEP  819
#define QSCALE 0.17677669529663687f   // 1/sqrt(32)

// workspace layout (float offsets)
#define WS_Q    0
#define WS_K    (WS_Q + NN*DD)
#define WS_V    (WS_K + MT*DD)
#define WS_REG  (WS_V + MT*DD)
#define WS_OATT (WS_REG + DD)
#define WS_GP   (WS_OATT + NN*DD)

// output layout (float offsets): final_out, raw_scores, gate_mean, attn_mean
#define O_FINAL 0
#define O_RAW   (NN*DD)
#define O_GMEAN (O_RAW + NN*MT)
#define O_AMEAN (O_GMEAN + 1)

__device__ __forceinline__ unsigned int fkey(float f) {
  // order-preserving map: float compare == unsigned compare on key
  unsigned int u = __float_as_uint(f);
  return (u & 0x80000000u) ? ~u : (u | 0x80000000u);
}

__device__ __forceinline__ v8f wmma_f32(v2f a, v2f b, v8f c) {
  // D = A(16x4) * B(4x16) + C(16x16), f32, wave32
  return __builtin_amdgcn_wmma_f32_16x16x4_f32(
      false, a, false, b, (short)0, c, false, false);
}

// ---------------------------------------------------------------------------
// regime_mod = sigmoid([z_p, z_s] @ W_gate^T + b_gate)   (1 x 128)
// ---------------------------------------------------------------------------
__global__ __launch_bounds__(128) void k_regime(
    const float* __restrict__ zp, const float* __restrict__ zs,
    const float* __restrict__ Wg, const float* __restrict__ bg,
    float* __restrict__ reg) {
  int j = threadIdx.x;
  float acc = bg[j];
  for (int i = 0; i < 128; ++i) acc += zp[i] * Wg[j * 256 + i];
  for (int i = 0; i < 128; ++i) acc += zs[i] * Wg[j * 256 + 128 + i];
  reg[j] = 1.0f / (1.0f + __expf(-acc));
}

// ---------------------------------------------------------------------------
// Projection GEMM: dst = rows @ W^T  (optionally * regime per column)
// mode 0: q (rows = stock, scale by regime); mode 1/2: k/v (rows = nodes)
// One WG = 16 rows x 128 cols; 8 waves, each wave owns one 16-col tile.
// ---------------------------------------------------------------------------
__global__ __launch_bounds__(256) void k_proj(
    const float* __restrict__ stock, const float* __restrict__ zp,
    const float* __restrict__ zs, const float* __restrict__ W,
    const float* __restrict__ regime, float* __restrict__ dst, int mode) {
  __shared__ float As[16 * 132];  // padded stride vs bank conflicts
  const int tid = threadIdx.x;
  const int tile = blockIdx.x;

  for (int idx = tid; idx < 2048; idx += 256) {
    int row = idx >> 7, col = idx & 127;
    int g = tile * 16 + row;
    float v = 0.0f;
    if (mode == 0) {
      v = stock[g * DD + col];
    } else {
      if (g == 0)        v = zp[col];
      else if (g == 1)   v = zs[col];
      else if (g < MT)   v = stock[(g - 2) * DD + col];
    }
    As[row * 132 + col] = v;
  }
  __syncthreads();

  const int wv = tid >> 5, lane = tid & 31, lm = lane & 15, hi = lane >> 4;
  const int col = wv * 16 + lm;

  v8f acc = {};
  for (int k0 = 0; k0 < DD; k0 += 4) {
    v2f a, b;
    a.x = As[lm * 132 + k0 + 2 * hi];
    a.y = As[lm * 132 + k0 + 2 * hi + 1];
    b.x = W[col * DD + k0 + 2 * hi];      // B[k][n] = W[n][k]
    b.y = W[col * DD + k0 + 2 * hi + 1];
    acc = wmma_f32(a, b, acc);
  }

  float rm = (mode == 0) ? regime[col] : 1.0f;
#pragma unroll
  for (int r = 0; r < 8; ++r) {
    int g = tile * 16 + r + 8 * hi;
    float val = acc[r] * rm;
    if (mode == 0)      dst[g * DD + col] = val;
    else if (g < MT)    dst[g * DD + col] = val;
  }
}

// ---------------------------------------------------------------------------
// Fused attention: scores (WMMA) -> raw mean -> exact top-k radix select ->
// masked softmax -> attn.mean -> attn @ v (WMMA).  One WG per 16 query rows.
// ---------------------------------------------------------------------------
__global__ __launch_bounds__(256) void k_attn(
    const float* __restrict__ qws, const float* __restrict__ kws,
    const float* __restrict__ vws, float* __restrict__ oatt,
    float* __restrict__ raw, float* __restrict__ amean) {
  __shared__ float sc[16 * SCS];          // 263232 B  score slab (one head)
  __shared__ float qh[16 * 36];           // A block, padded stride
  __shared__ float wred[8 * 256];         // cross-wave GEMM reduce
  __shared__ float red[16 * 17];          // per-row reductions
  __shared__ unsigned int hist[256];      // radix histogram
  __shared__ unsigned int thrK[16];       // per-row threshold key
  __shared__ float rowmax[16], denomS[16];
  __shared__ int sB, sW;

  const int tid = threadIdx.x;
  const int r0 = blockIdx.x * 16;
  const int wv = tid >> 5, lane = tid & 31, lm = lane & 15, hi = lane >> 4;
  const int row16 = tid >> 4, t16 = tid & 15;

  for (int h = 0; h < HH; ++h) {
    // ---- stage q block (16 rows x 32 head-cols) ----
    for (int i = tid; i < 512; i += 256) {
      int r = i >> 5, c = i & 31;
      qh[r * 36 + c] = qws[(r0 + r) * DD + h * DHD + c];
    }
    __syncthreads();

    // ---- scores = q_h (16x32) @ k_h^T (32xM), WMMA f32 16x16x4 ----
    for (int t = wv; t < 257; t += 8) {
      int m0 = t * 16;
      int km = m0 + lm;                 // key index for this lane's B column
      bool kv = (km < MT);
      const float* kb = kws + (long)km * DD + h * DHD;
      v8f acc = {};
      for (int k0 = 0; k0 < DHD; k0 += 4) {
        v2f a, b;
        a.x = qh[lm * 36 + k0 + 2 * hi];
        a.y = qh[lm * 36 + k0 + 2 * hi + 1];
        b.x = kv ? kb[k0 + 2 * hi] : 0.0f;
        b.y = kv ? kb[k0 + 2 * hi + 1] : 0.0f;
        acc = wmma_f32(a, b, acc);
      }
#pragma unroll
      for (int r = 0; r < 8; ++r) {
        int qr = r + 8 * hi;
        sc[qr * SCS + km] = kv ? acc[r] * QSCALE : -1.0e30f;
      }
    }
    __syncthreads();

    // ---- raw_scores accumulation (pre-mask), L2-local RMW ----
    for (int r = 0; r < 16; ++r) {
      float* g = raw + (long)(r0 + r) * MT;
      for (int m = tid; m < MT; m += 256) {
        float v = sc[r * SCS + m] * 0.25f;
        if (h == 0) g[m] = v; else g[m] += v;
      }
    }
    __syncthreads();

    // ---- exact 819th-largest per row: 4-pass 8-bit radix select ----
    for (int r = 0; r < 16; ++r) {
      unsigned int prefix = 0;
      int want = KKEEP;
      for (int pass = 0; pass < 4; ++pass) {
        int shift = 24 - 8 * pass;
        hist[tid] = 0u;
        __syncthreads();
        for (int i = tid; i < NN; i += 256) {
          unsigned int u = fkey(sc[r * SCS + 2 + i]);   // stock scores only
          bool in = (pass == 0) || ((u >> (shift + 8)) == prefix);
          if (in) atomicAdd(&hist[(u >> shift) & 255u], 1u);
        }
        __syncthreads();
        if (tid == 0) {
          int w = want, b = 255;
          for (; b > 0; --b) {
            int c = (int)hist[b];
            if (w <= c) break;
            w -= c;
          }
          sB = b; sW = w;
        }
        __syncthreads();
        prefix = (prefix << 8) | (unsigned int)sB;
        want = sW;
        __syncthreads();
      }
      if (tid == 0) thrK[r] = prefix;   // exact key of 819th largest
      __syncthreads();
    }

    // ---- mask (stocks only) + row max ----
    {
      float mx = -3.0e38f;
      for (int c = t16; c < MT; c += 16) {
        float v = sc[row16 * SCS + c];
        if (c >= 2 && fkey(v) < thrK[row16]) {
          v = -1.0e9f;
          sc[row16 * SCS + c] = v;
        }
        mx = fmaxf(mx, v);
      }
      red[row16 * 17 + t16] = mx;
    }
    __syncthreads();
    if (t16 == 0) {
      float mx = red[row16 * 17];
      for (int i = 1; i < 16; ++i) mx = fmaxf(mx, red[row16 * 17 + i]);
      rowmax[row16] = mx;
    }
    __syncthreads();

    // ---- exp + denom (normalization deferred), zero pad columns ----
    {
      float s = 0.0f, mx = rowmax[row16];
      for (int c = t16; c < MP; c += 16) {
        if (c < MT) {
          float e = __expf(sc[row16 * SCS + c] - mx);
          sc[row16 * SCS + c] = e;
          s += e;
        } else {
          sc[row16 * SCS + c] = 0.0f;
        }
      }
      red[row16 * 17 + t16] = s;
    }
    __syncthreads();
    if (t16 == 0) {
      float s = 0.0f;
      for (int i = 0; i < 16; ++i) s += red[row16 * 17 + i];
      denomS[row16] = s;
    }
    __syncthreads();

    // ---- attn.mean accumulation (L2-local RMW) ----
    for (int r = 0; r < 16; ++r) {
      float inv = 0.25f / denomS[r];
      float* g = amean + (long)(r0 + r) * MT;
      for (int m = tid; m < MT; m += 256) {
        float a = sc[r * SCS + m] * inv;
        if (h == 0) g[m] = a; else g[m] += a;
      }
    }

    // ---- out_h = attn(16xM) @ v_h(Mx32), WMMA, K split over 4 wave pairs --
    {
      int qtr = wv >> 1, ct = wv & 1;
      int col = h * DHD + ct * 16 + lm;
      int kbeg = qtr * 1028, kend = kbeg + 1028;   // 4112/4 quarters
      v8f acc = {};
      for (int k0 = kbeg; k0 < kend; k0 += 4) {
        int kr0 = k0 + 2 * hi, kr1 = kr0 + 1;
        v2f a, b;
        a.x = sc[lm * SCS + kr0];
        a.y = sc[lm * SCS + kr1];
        b.x = (kr0 < MT) ? vws[(long)kr0 * DD + col] : 0.0f;
        b.y = (kr1 < MT) ? vws[(long)kr1 * DD + col] : 0.0f;
        acc = wmma_f32(a, b, acc);
      }
#pragma unroll
      for (int r = 0; r < 8; ++r)
        wred[wv * 256 + (r + 8 * hi) * 16 + lm] = acc[r];
    }
    __syncthreads();
    for (int e = tid; e < 512; e += 256) {
      int r = e >> 5, c = e & 31;
      int ct = c >> 4, cl = c & 15;
      float s = wred[(ct) * 256 + r * 16 + cl] +
                wred[(2 + ct) * 256 + r * 16 + cl] +
                wred[(4 + ct) * 256 + r * 16 + cl] +
                wred[(6 + ct) * 256 + r * 16 + cl];
      oatt[(long)(r0 + r) * DD + h * DHD + c] = s / denomS[r];
    }
    __syncthreads();
  }
}

// ---------------------------------------------------------------------------
// Output projection + adaptive gate + RMS norm; per-row gate partial sums.
// ---------------------------------------------------------------------------
__global__ __launch_bounds__(128) void k_post(
    const float* __restrict__ stock, const float* __restrict__ oatt,
    const float* __restrict__ Wout, const float* __restrict__ bout,
    const float* __restrict__ Wag, const float* __restrict__ bag,
    const float* __restrict__ rmsw, float* __restrict__ finalOut,
    float* __restrict__ gp) {
  __shared__ float srow[128], orow[128], prow[128], gred[128];
  int n = blockIdx.x, t = threadIdx.x;
  srow[t] = stock[n * DD + t];
  orow[t] = oatt[n * DD + t];
  __syncthreads();

  float p = bout[t];
  for (int i = 0; i < DD; ++i) p += orow[i] * Wout[t * DD + i];
  prow[t] = p;
  __syncthreads();

  float g = bag[t];
  for (int i = 0; i < DD; ++i) g += srow[i] * Wag[t * 256 + i];
  for (int i = 0; i < DD; ++i) g += prow[i] * Wag[t * 256 + 128 + i];
  float gate = 1.0f / (1.0f + __expf(-g));
  float gated = srow[t] + gate * prow[t];

  gred[t] = gated * gated;
  __syncthreads();
  for (int s = 64; s > 0; s >>= 1) {
    if (t < s) gred[t] += gred[t + s];
    __syncthreads();
  }
  float rms = rsqrtf(gred[0] / 128.0f + 1e-6f);
  finalOut[n * DD + t] = gated * rms * rmsw[t];
  __syncthreads();

  gred[t] = gate;
  __syncthreads();
  for (int s = 64; s > 0; s >>= 1) {
    if (t < s) gred[t] += gred[t + s];
    __syncthreads();
  }
  if (t == 0) gp[n] = gred[0];
}

// ---------------------------------------------------------------------------
// gate.mean(): fixed-order deterministic reduction of per-row partials.
// ---------------------------------------------------------------------------
__global__ __launch_bounds__(256) void k_gmean(const float* __restrict__ gp,
                                               float* __restrict__ out) {
  __shared__ float sr[256];
  int t = threadIdx.x;
  float s = 0.0f;
  for (int i = t; i < NN; i += 256) s += gp[i];
  sr[t] = s;
  __syncthreads();
  for (int st = 128; st > 0; st >>= 1) {
    if (t < st) sr[t] += sr[t + st];
    __syncthreads();
  }
  if (t == 0) out[0] = sr[0] / (float)(NN * DD);
}

// ---------------------------------------------------------------------------
extern "C" void kernel_launch(void* const* d_in, const int* in_sizes, int n_in,
                              void* d_out, int out_size, void* d_ws,
                              size_t ws_size, hipStream_t stream) {
  (void)in_sizes; (void)n_in; (void)out_size; (void)ws_size;
  const float* stock = (const float*)d_in[0];
  const float* zp    = (const float*)d_in[1];
  const float* zs    = (const float*)d_in[2];
  const float* Wq    = (const float*)d_in[3];
  const float* Wk    = (const float*)d_in[4];
  const float* Wv    = (const float*)d_in[5];
  const float* Wg    = (const float*)d_in[6];
  const float* bg    = (const float*)d_in[7];
  const float* Wag   = (const float*)d_in[8];
  const float* bag   = (const float*)d_in[9];
  const float* Wout  = (const float*)d_in[10];
  const float* bout  = (const float*)d_in[11];
  const float* rmsw  = (const float*)d_in[12];
  float* out = (float*)d_out;
  float* ws  = (float*)d_ws;

  k_regime<<<1, 128, 0, stream>>>(zp, zs, Wg, bg, ws + WS_REG);
  k_proj<<<256, 256, 0, stream>>>(stock, zp, zs, Wq, ws + WS_REG, ws + WS_Q, 0);
  k_proj<<<257, 256, 0, stream>>>(stock, zp, zs, Wk, ws + WS_REG, ws + WS_K, 1);
  k_proj<<<257, 256, 0, stream>>>(stock, zp, zs, Wv, ws + WS_REG, ws + WS_V, 2);
  k_attn<<<256, 256, 0, stream>>>(ws + WS_Q, ws + WS_K, ws + WS_V, ws + WS_OATT,
                                  out + O_RAW, out + O_AMEAN);
  k_post<<<4096, 128, 0, stream>>>(stock, ws + WS_OATT, Wout, bout, Wag, bag,
                                   rmsw, out + O_FINAL, ws + WS_GP);
  k_gmean<<<1, 256, 0, stream>>>(ws + WS_GP, out + O_GMEAN);
}